// WinMSA_3109556322814
// MI455X (gfx1250) — compile-verified
//
#include <hip/hip_runtime.h>
#include <cstddef>

typedef __attribute__((ext_vector_type(2))) float v2f;
typedef __attribute__((ext_vector_type(4))) float v4f;
typedef __attribute__((ext_vector_type(8))) float v8f;
typedef int i128v __attribute__((vector_size(16)));   // 4 x i32, matches builtin param

#define HEADS 12
#define EMB   384
#define ATT   64
#define NBLK  (32 * 49)
#define SCALE 0.17677669529663687f /* 32^-0.5 */

// LDS layout (floats)
#define XS_OFF 0        // 64 x 388
#define XS_ST  388
#define QS_OFF 24832    // 64 x 36  (reused as Oh after attention)
#define QS_ST  36
#define KS_OFF 27136    // 64 x 36
#define KS_ST  36
#define VS_OFF 29440    // 64 x 40
#define VS_ST  40
#define PS_OFF 32000    // 64 x 68  (unioned with W-chunk 32 x 112)
#define PS_ST  68
#define WS_ST  112
#define LDS_FLOATS 36352

#define GLOBAL_AS __attribute__((address_space(1)))
#define LDS_AS    __attribute__((address_space(3)))

#if __has_builtin(__builtin_amdgcn_global_load_async_to_lds_b128)
#define HAVE_ASYNC_LDS 1
#else
#define HAVE_ASYNC_LDS 0
#endif

static __device__ __forceinline__ void async_wait0() {
#if __has_builtin(__builtin_amdgcn_s_wait_asynccnt)
  __builtin_amdgcn_s_wait_asynccnt(0);
#else
  asm volatile("s_wait_asynccnt 0" ::: "memory");
#endif
}

#if HAVE_ASYNC_LDS
static __device__ __forceinline__ void async_cp16(const float* g, float* l) {
  __builtin_amdgcn_global_load_async_to_lds_b128((GLOBAL_AS i128v*)g,
                                                 (LDS_AS i128v*)l, 0, 0);
}
#endif

static __device__ __forceinline__ v8f wmma4(v2f a, v2f b, v8f c) {
  return __builtin_amdgcn_wmma_f32_16x16x4_f32(false, a, false, b, (short)0, c,
                                               false, false);
}

// A fragment: 16x4 tile at (r0, k0). lanes 0-15: K=k0,k0+1 ; lanes 16-31: K=k0+2,k0+3
static __device__ __forceinline__ v2f ldsA(const float* S, int st, int r0,
                                           int k0, int l15, int hl) {
  const float* p = S + (r0 + l15) * st + k0 + 2 * hl;
  v2f a; a.x = p[0]; a.y = p[1]; return a;
}
// B fragment: 4x16 tile at (k0, n0), B stored row-major S[k][n]
static __device__ __forceinline__ v2f ldsB(const float* S, int st, int k0,
                                           int n0, int l15, int hl) {
  const float* p = S + (k0 + 2 * hl) * st + n0 + l15;
  v2f b; b.x = p[0]; b.y = p[st]; return b;
}
static __device__ __forceinline__ v2f glbB(const float* W, int ldw, int k0,
                                           int n0, int l15, int hl) {
  const float* p = W + (size_t)(k0 + 2 * hl) * ldw + n0 + l15;
  v2f b; b.x = p[0]; b.y = p[ldw]; return b;
}
static __device__ __forceinline__ v8f vzero() {
  v8f z = {0.f, 0.f, 0.f, 0.f, 0.f, 0.f, 0.f, 0.f};
  return z;
}

__global__ __launch_bounds__(256, 1) void winmsa_kernel(
    const float* __restrict__ X, const float* __restrict__ Wqkv,
    const float* __restrict__ PB, const float* __restrict__ Wo,
    const float* __restrict__ bo, const int* __restrict__ relp,
    float* __restrict__ Y) {
  extern __shared__ float sm[];
  float* Xs = sm + XS_OFF;
  float* Qs = sm + QS_OFF;   // later reused as Oh
  float* Ks = sm + KS_OFF;
  float* Vs = sm + VS_OFF;
  float* Ps = sm + PS_OFF;   // unioned with W-chunk staging
  float* Ws = sm + PS_OFF;

  const int t    = threadIdx.x;
  const int wv   = t >> 5;
  const int lane = t & 31;
  const int hl   = lane >> 4;
  const int l15  = lane & 15;

  const float* Xw = X + (size_t)blockIdx.x * (ATT * EMB);
  float*       Yw = Y + (size_t)blockIdx.x * (ATT * EMB);

  // ---- stage X window (64 x 384) into LDS: async global->LDS, no VGPR bounce ----
#if HAVE_ASYNC_LDS
#pragma unroll 4
  for (int i = 0; i < 24; ++i) {
    int idx4 = t + i * 256;          // 0..6143 float4s
    int r = idx4 / 96, c4 = idx4 % 96;
    async_cp16(Xw + (size_t)r * EMB + c4 * 4, Xs + r * XS_ST + c4 * 4);
  }
  async_wait0();
#else
#pragma unroll 4
  for (int i = 0; i < 24; ++i) {
    int idx4 = t + i * 256;
    int r = idx4 / 96, c4 = idx4 % 96;
    v4f xv = *(const v4f*)(Xw + (size_t)r * EMB + c4 * 4);
    float* d = Xs + r * XS_ST + c4 * 4;
    d[0] = xv.x; d[1] = xv.y; d[2] = xv.z; d[3] = xv.w;
  }
#endif

  // Y accumulators held in registers: wave owns N-tiles {3w..3w+2} x M-tiles 0..3
  v8f yacc[3][4];
  for (int i = 0; i < 3; ++i)
    for (int m = 0; m < 4; ++m) yacc[i][m] = vzero();

  __syncthreads();

  for (int h = 0; h < HEADS; ++h) {
    // ================= QKV: (64x384) @ (384x96) =================
    v8f acc[3];
    acc[0] = vzero(); acc[1] = vzero(); acc[2] = vzero();
    const int tile0 = wv * 3;  // 24 tiles = 4 M x 6 N ; 3 per wave
    for (int kc = 0; kc < 12; ++kc) {
      // stage 32 K-rows of this head's 96 qkv columns (3 x 32-float segments/row)
#if HAVE_ASYNC_LDS
#pragma unroll
      for (int i = 0; i < 3; ++i) {
        int e = t + i * 256;         // 0..767 b128 transfers
        int f4 = e & 7;              // float4 within 32-float segment
        int rp = e >> 3;             // 0..95
        int p = rp % 3, r = rp / 3;
        async_cp16(Wqkv + (size_t)(kc * 32 + r) * (3 * EMB) + p * EMB +
                       h * 32 + f4 * 4,
                   Ws + r * WS_ST + p * 32 + f4 * 4);
      }
      async_wait0();
#else
#pragma unroll 4
      for (int i = 0; i < 12; ++i) {
        int e = t + i * 256;         // 0..3071
        int r = e / 96, c = e % 96;
        int p = c >> 5, d = c & 31;
        Ws[r * WS_ST + c] =
            Wqkv[(size_t)(kc * 32 + r) * (3 * EMB) + p * EMB + h * 32 + d];
      }
#endif
      __syncthreads();
      for (int tt = 0; tt < 3; ++tt) {
        int tile = tile0 + tt;
        int mt = tile & 3, nt = tile >> 2;
#pragma unroll
        for (int kk = 0; kk < 8; ++kk) {
          v2f a = ldsA(Xs, XS_ST, mt * 16, kc * 32 + kk * 4, l15, hl);
          v2f b = ldsB(Ws, WS_ST, kk * 4, nt * 16, l15, hl);
          acc[tt] = wmma4(a, b, acc[tt]);
        }
      }
      __syncthreads();
    }
    // scatter Q/K/V tiles to LDS
    for (int tt = 0; tt < 3; ++tt) {
      int tile = tile0 + tt;
      int mt = tile & 3, nt = tile >> 2;
      int p = nt >> 1, d0 = (nt & 1) * 16;
      float* B = (p == 0) ? Qs : ((p == 1) ? Ks : Vs);
      int st = (p == 2) ? VS_ST : QS_ST;
#pragma unroll
      for (int j = 0; j < 8; ++j)
        B[(mt * 16 + j + 8 * hl) * st + d0 + l15] = acc[tt][j];
    }
    __syncthreads();

    // ================= attn = Q @ K^T * scale + bias =================
    {
      const int a0 = wv * 2;         // 16 tiles, 2 per wave (same q-tile)
      const int qt = a0 >> 2;
      v8f at0 = vzero(), at1 = vzero();
#pragma unroll
      for (int kk = 0; kk < 8; ++kk) {
        v2f a = ldsA(Qs, QS_ST, qt * 16, kk * 4, l15, hl);
        v2f b0 = ldsA(Ks, KS_ST, (a0 & 3) * 16, kk * 4, l15, hl);       // B = K^T
        v2f b1 = ldsA(Ks, KS_ST, ((a0 + 1) & 3) * 16, kk * 4, l15, hl);
        at0 = wmma4(a, b0, at0);
        at1 = wmma4(a, b1, at1);
      }
      const float* pbh = PB + (size_t)h * (127 * 127);
      for (int u = 0; u < 2; ++u) {
        v8f av = (u == 0) ? at0 : at1;
        int kcol = ((a0 + u) & 3) * 16 + l15;
#pragma unroll
        for (int j = 0; j < 8; ++j) {
          int q = qt * 16 + j + 8 * hl;
          int idx = relp[q * ATT + kcol];
          Ps[q * PS_ST + kcol] = av[j] * SCALE + pbh[idx];
        }
      }
    }
    __syncthreads();

    // ================= softmax over rows of 64 =================
    if (t < ATT) {
      float* row = Ps + t * PS_ST;
      float m = row[0];
      for (int c = 1; c < ATT; ++c) m = fmaxf(m, row[c]);
      float s = 0.f;
      for (int c = 0; c < ATT; ++c) { float e = __expf(row[c] - m); row[c] = e; s += e; }
      float inv = 1.f / s;
      for (int c = 0; c < ATT; ++c) row[c] *= inv;
    }
    __syncthreads();

    // ================= Oh = P @ V  (64x64)@(64x32), 1 tile/wave =================
    {
      int mt = wv >> 1, nt = wv & 1;
      v8f o = vzero();
#pragma unroll
      for (int kk = 0; kk < 16; ++kk) {
        v2f a = ldsA(Ps, PS_ST, mt * 16, kk * 4, l15, hl);
        v2f b = ldsB(Vs, VS_ST, kk * 4, nt * 16, l15, hl);
        o = wmma4(a, b, o);
      }
      float* Oh = Qs;  // Q no longer needed
#pragma unroll
      for (int j = 0; j < 8; ++j)
        Oh[(mt * 16 + j + 8 * hl) * QS_ST + nt * 16 + l15] = o[j];
    }
    __syncthreads();

    // ================= Y += Oh @ Wo[h*32:(h+1)*32, :] =================
    {
      const float* Oh = Qs;
      for (int i = 0; i < 3; ++i) {
        int nt = wv * 3 + i;
#pragma unroll
        for (int kk = 0; kk < 8; ++kk) {
          v2f b = glbB(Wo, EMB, h * 32 + kk * 4, nt * 16, l15, hl);
#pragma unroll
          for (int mt = 0; mt < 4; ++mt) {
            v2f a = ldsA(Oh, QS_ST, mt * 16, kk * 4, l15, hl);
            yacc[i][mt] = wmma4(a, b, yacc[i][mt]);
          }
        }
      }
    }
    __syncthreads();
  }

  // ---- write Y + b_o ----
  for (int i = 0; i < 3; ++i) {
    int col = (wv * 3 + i) * 16 + l15;
    float bb = bo[col];
    for (int mt = 0; mt < 4; ++mt) {
#pragma unroll
      for (int j = 0; j < 8; ++j) {
        int r = mt * 16 + j + 8 * hl;
        Yw[(size_t)r * EMB + col] = yacc[i][mt][j] + bb;
      }
    }
  }
}

extern "C" void kernel_launch(void* const* d_in, const int* in_sizes, int n_in,
                              void* d_out, int out_size, void* d_ws,
                              size_t ws_size, hipStream_t stream) {
  (void)in_sizes; (void)n_in; (void)out_size; (void)d_ws; (void)ws_size;
  const float* X    = (const float*)d_in[0];
  const float* Wqkv = (const float*)d_in[1];
  const float* PB   = (const float*)d_in[2];
  const float* Wo   = (const float*)d_in[3];
  const float* bo   = (const float*)d_in[4];
  const int*   relp = (const int*)d_in[5];
  float* Y = (float*)d_out;

  const size_t ldsBytes = (size_t)LDS_FLOATS * sizeof(float); // ~142 KB
  winmsa_kernel<<<dim3(NBLK), dim3(256), ldsBytes, stream>>>(X, Wqkv, PB, Wo,
                                                             bo, relp, Y);
}